// Attention_50964081935360
// MI455X (gfx1250) — compile-verified
//
#include <hip/hip_runtime.h>

// CDNA5 / gfx1250, wave32.
typedef __attribute__((ext_vector_type(2))) float v2f;
typedef __attribute__((ext_vector_type(8))) float v8f;

#define B_  32
#define S_  4096
#define D_  1024
#define TILES_PER_B   (S_ / 16)                         // 256
#define WAVES_PER_BLK 8
#define K1_BLOCKS     (B_ * TILES_PER_B / WAVES_PER_BLK) // 1024
#define K2_THREADS    256
#define PER_T         (S_ / K2_THREADS)                  // 16

// ---------------------------------------------------------------------------
// Kernel 1: single fused pass over key (512 MB -> the only HBM-heavy pass).
// Each wave owns a 16-row sequence tile of one batch element and accumulates
//   scores[b, s0..s0+15] = key_tile(16 x 1024) * query[b](1024)
// via v_wmma_f32_16x16x4_f32 with the query chunk broadcast across M rows,
// while simultaneously accumulating sum(key^2) from the same loaded data.
//
// Operand layouts (wave32, from CDNA5 ISA 7.12.2):
//   A (16x4 f32): lanes 0-15 -> K=0,1 ; lanes 16-31 -> K=2,3   (M = lane%16)
//   B (4x16 f32): lanes 0-15 -> K=0,1 ; lanes 16-31 -> K=2,3   (N = lane%16)
//   C/D row M=0 lives in acc[0], lanes 0-15 (N = lane).
// ---------------------------------------------------------------------------
__global__ __launch_bounds__(256) void attn_scores_sumsq_kernel(
    const float* __restrict__ key, const float* __restrict__ query,
    float* __restrict__ scores, float* __restrict__ partials)
{
  __shared__ float red[WAVES_PER_BLK];
  const int lane = threadIdx.x & 31;
  const int wave = threadIdx.x >> 5;
  const int tile = blockIdx.x * WAVES_PER_BLK + wave;   // [0, 8192)
  const int b    = tile / TILES_PER_B;
  const int s0   = (tile % TILES_PER_B) * 16;
  const bool lo  = lane < 16;
  const int row  = s0 + (lane & 15);

  const float* __restrict__ krow = key   + ((size_t)b * S_ + row) * D_;
  const float* __restrict__ q    = query + (size_t)b * D_;

  v8f acc = {};       // f32 16x16 accumulator (8 VGPRs)
  float ss = 0.f;     // sum of squares (each float4 is loaded by 2 lanes -> 2x)

  #pragma unroll 4
  for (int k0 = 0; k0 < D_; k0 += 4) {
    const float4 kv = *(const float4*)(krow + k0);  // 16B contiguous per lane
    const float4 qv = *(const float4*)(q + k0);     // uniform, L0/L2 resident

    v2f a, bmat;
    a[0]    = lo ? qv.x : qv.z;  a[1]    = lo ? qv.y : qv.w;
    bmat[0] = lo ? kv.x : kv.z;  bmat[1] = lo ? kv.y : kv.w;

    acc = __builtin_amdgcn_wmma_f32_16x16x4_f32(
        /*neg_a=*/false, a, /*neg_b=*/false, bmat,
        /*c_mod=*/(short)0, acc, /*reuse_a=*/false, /*reuse_b=*/false);

    ss = fmaf(kv.x, kv.x, ss);
    ss = fmaf(kv.y, kv.y, ss);
    ss = fmaf(kv.z, kv.z, ss);
    ss = fmaf(kv.w, kv.w, ss);
  }

  // D row M=0: lanes 0..15 hold scores for key rows s0+0..s0+15.
  if (lo) scores[(size_t)b * S_ + s0 + lane] = acc[0];

  // wave32 reduction of sum-of-squares, then block reduction -> one partial.
  for (int off = 16; off > 0; off >>= 1)
    ss += __shfl_xor(ss, off, 32);
  if (lane == 0) red[wave] = ss;
  __syncthreads();
  if (threadIdx.x == 0) {
    float t = 0.f;
    #pragma unroll
    for (int w = 0; w < WAVES_PER_BLK; ++w) t += red[w];
    partials[blockIdx.x] = t;   // NOTE: 2x the true sum (dual-lane loads)
  }
}

// ---------------------------------------------------------------------------
// Kernel 2: one block per batch row. Deterministically re-reduces the 1024
// partials (tiny, L2-resident), then masked softmax in-place on d_out.
// softmax-then-mask-then-renorm == masked softmax (Z cancels).
// ---------------------------------------------------------------------------
__global__ __launch_bounds__(K2_THREADS) void attn_softmax_kernel(
    float* __restrict__ out, const float* __restrict__ partials,
    const int* __restrict__ seq_lens)
{
  __shared__ float sred[K2_THREADS];
  const int t = threadIdx.x;
  const int b = blockIdx.x;

  // Frobenius norm: partials hold 2x the sum of squares.
  float ps = 0.f;
  for (int i = t; i < K1_BLOCKS; i += K2_THREADS) ps += partials[i];
  sred[t] = ps; __syncthreads();
  for (int s = K2_THREADS / 2; s > 0; s >>= 1) {
    if (t < s) sred[t] += sred[t + s];
    __syncthreads();
  }
  const float inv_denom = 1.0f / sqrtf(sred[0] * 0.5f);
  __syncthreads();

  const int len = seq_lens[b];
  float v[PER_T];
  float m = -__builtin_inff();
  #pragma unroll
  for (int j = 0; j < PER_T; ++j) {
    const int i = j * K2_THREADS + t;
    const float s = out[(size_t)b * S_ + i] * inv_denom;
    v[j] = s;
    if (i < len) m = fmaxf(m, s);
  }
  sred[t] = m; __syncthreads();
  for (int s = K2_THREADS / 2; s > 0; s >>= 1) {
    if (t < s) sred[t] = fmaxf(sred[t], sred[t + s]);
    __syncthreads();
  }
  m = sred[0];
  __syncthreads();

  float lsum = 0.f;
  #pragma unroll
  for (int j = 0; j < PER_T; ++j) {
    const int i = j * K2_THREADS + t;
    const float e = (i < len) ? expf(v[j] - m) : 0.f;
    v[j] = e;
    lsum += e;
  }
  sred[t] = lsum; __syncthreads();
  for (int s = K2_THREADS / 2; s > 0; s >>= 1) {
    if (t < s) sred[t] += sred[t + s];
    __syncthreads();
  }
  const float inv_sum = 1.0f / sred[0];

  #pragma unroll
  for (int j = 0; j < PER_T; ++j) {
    const int i = j * K2_THREADS + t;
    out[(size_t)b * S_ + i] = v[j] * inv_sum + 1e-15f;  // masked -> exactly 1e-15
  }
}

extern "C" void kernel_launch(void* const* d_in, const int* in_sizes, int n_in,
                              void* d_out, int out_size, void* d_ws, size_t ws_size,
                              hipStream_t stream) {
  const float* key      = (const float*)d_in[0];  // [32, 4096, 1024] f32
  const float* query    = (const float*)d_in[1];  // [32, 1024] f32
  const int*   seq_lens = (const int*)d_in[2];    // [32] i32
  float* out      = (float*)d_out;                // [32, 4096, 1] f32 (scores staged here)
  float* partials = (float*)d_ws;                 // K1_BLOCKS floats (4 KB)

  attn_scores_sumsq_kernel<<<K1_BLOCKS, 256, 0, stream>>>(key, query, out, partials);
  attn_softmax_kernel<<<B_, K2_THREADS, 0, stream>>>(out, partials, seq_lens);
}